// ChebyshevConvolution_36756330119384
// MI455X (gfx1250) — compile-verified
//
#include <hip/hip_runtime.h>
#include <hip/hip_bf16.h>

typedef __attribute__((ext_vector_type(2))) float v2f;
typedef __attribute__((ext_vector_type(8))) float v8f;

static constexpr int F_INC  = 165;
static constexpr int F_PADC = 168;   // padded to multiple of 4 for WMMA K-steps
static constexpr int HIDC   = 128;
static constexpr int NCLS   = 2;

// ---------------- elementwise helpers ----------------
__global__ void k_zero(float* __restrict__ p, long long n) {
  long long i = (long long)blockIdx.x * blockDim.x + threadIdx.x;
  long long s = (long long)gridDim.x * blockDim.x;
  for (; i < n; i += s) p[i] = 0.0f;
}

__global__ void k_negcopy(float* __restrict__ dst, const float* __restrict__ src, long long n) {
  long long i = (long long)blockIdx.x * blockDim.x + threadIdx.x;
  long long s = (long long)gridDim.x * blockDim.x;
  for (; i < n; i += s) dst[i] = -src[i];
}

__global__ void k_bias_act(float* __restrict__ o, const float* __restrict__ b,
                           long long n, int h, int relu6) {
  long long i = (long long)blockIdx.x * blockDim.x + threadIdx.x;
  long long s = (long long)gridDim.x * blockDim.x;
  for (; i < n; i += s) {
    float v = o[i] + b[(int)(i % h)];
    if (relu6) v = fminf(fmaxf(v, 0.0f), 6.0f);
    o[i] = v;
  }
}

// ---------------- graph normalization ----------------
__global__ void k_deg(const long long* __restrict__ row, float* __restrict__ deg, int E) {
  int e = blockIdx.x * blockDim.x + threadIdx.x;
  if (e < E) unsafeAtomicAdd(&deg[(int)row[e]], 1.0f);
}

__global__ void k_dinv(const float* __restrict__ deg, float* __restrict__ dinv, int n) {
  int i = blockIdx.x * blockDim.x + threadIdx.x;
  if (i < n) {
    float d = deg[i];
    dinv[i] = (d > 0.0f) ? rsqrtf(fmaxf(d, 1.0f)) : 0.0f;
  }
}

__global__ void k_norm(const long long* __restrict__ row, const long long* __restrict__ col,
                       const float* __restrict__ dinv, float* __restrict__ nrm, int E) {
  int e = blockIdx.x * blockDim.x + threadIdx.x;
  if (e < E) nrm[e] = -dinv[(int)row[e]] * dinv[(int)col[e]];
}

// ---------------- pad copies (K -> multiple of 4) ----------------
__global__ void k_pad_x(const float* __restrict__ x, float* __restrict__ xp, int n) {
  long long total = (long long)n * F_PADC;
  long long i = (long long)blockIdx.x * blockDim.x + threadIdx.x;
  long long s = (long long)gridDim.x * blockDim.x;
  for (; i < total; i += s) {
    int r = (int)(i / F_PADC), f = (int)(i % F_PADC);
    xp[i] = (f < F_INC) ? x[(long long)r * F_INC + f] : 0.0f;
  }
}

__global__ void k_pad_w(const float* __restrict__ w, float* __restrict__ wp) {
  long long total = 3LL * F_PADC * HIDC;
  long long i = (long long)blockIdx.x * blockDim.x + threadIdx.x;
  long long s = (long long)gridDim.x * blockDim.x;
  for (; i < total; i += s) {
    int k = (int)(i / (F_PADC * HIDC));
    int r = (int)((i / HIDC) % F_PADC);
    int c = (int)(i % HIDC);
    wp[i] = (r < F_INC) ? w[((long long)k * F_INC + r) * HIDC + c] : 0.0f;
  }
}

// ---------------- sparse propagation: out[row] += scale*norm*h[col] ----------------
__global__ void k_prop(float* __restrict__ out, const float* __restrict__ h,
                       const long long* __restrict__ row, const long long* __restrict__ col,
                       const float* __restrict__ nrm, int E, int F, int ld, float scale) {
  const int EPB = 8;
  int e0 = blockIdx.x * EPB;
  int e1 = e0 + EPB; if (e1 > E) e1 = E;
  for (int e = e0; e < e1; ++e) {
    int r = (int)row[e], c = (int)col[e];
    float w = scale * nrm[e];
    const float* __restrict__ src = h + (long long)c * ld;
    float* __restrict__ dst = out + (long long)r * ld;
    for (int f = threadIdx.x; f < F; f += blockDim.x)
      unsafeAtomicAdd(dst + f, w * src[f]);
  }
}

// ---------------- WMMA fp32 GEMM: C[M x 128] += A[M x K] * B[K x 128] ----------------
// One wave per 16x16 output tile; V_WMMA_F32_16X16X4_F32 over the K dimension.
__global__ void __launch_bounds__(32)
k_wmma_gemm(float* __restrict__ C, const float* __restrict__ A, const float* __restrict__ B,
            int M, int K) {
  const int lane = threadIdx.x;
  const int half = lane >> 4;     // 0: lanes 0-15, 1: lanes 16-31
  const int m    = lane & 15;
  const int row0 = blockIdx.x << 4;
  const int col0 = blockIdx.y << 4;

  // C/D layout: VGPR v holds row (v + 8*half), column = lane&15
  v8f c;
#pragma unroll
  for (int v = 0; v < 8; ++v) {
    int r = row0 + v + 8 * half;
    int rc = (r < M) ? r : (M - 1);
    c[v] = C[(long long)rc * HIDC + col0 + m];
  }

  // A layout: lane holds row (row0 + m), VGPR v holds K = v + 2*half (per 4-step)
  int ra = row0 + m; if (ra >= M) ra = M - 1;
  const float* __restrict__ arow = A + (long long)ra * K + 2 * half;
  // B layout: VGPR v holds K-row (v + 2*half), column = lane&15
  const float* __restrict__ bp = B + (long long)(2 * half) * HIDC + col0 + m;

  for (int kk = 0; kk < K; kk += 4) {
    v2f a = *(const v2f*)(arow + kk);          // global_load_b64, aligned (kk,half even)
    v2f b;
    b[0] = bp[(long long)kk * HIDC];
    b[1] = bp[(long long)kk * HIDC + HIDC];
    c = __builtin_amdgcn_wmma_f32_16x16x4_f32(
        /*neg_a=*/false, a, /*neg_b=*/false, b,
        /*c_mod=*/(short)0, c, /*reuse_a=*/false, /*reuse_b=*/false);
  }

#pragma unroll
  for (int v = 0; v < 8; ++v) {
    int r = row0 + v + 8 * half;
    if (r < M) C[(long long)r * HIDC + col0 + m] = c[v];
  }
}

// ---------------- tiny GEMM for final layer (Cout = 2) ----------------
__global__ void k_gemm2(float* __restrict__ C, const float* __restrict__ A,
                        const float* __restrict__ W, int n, int K, int accum) {
  int i = blockIdx.x * blockDim.x + threadIdx.x;
  if (i >= n) return;
  float a0 = 0.0f, a1 = 0.0f;
  const float* __restrict__ arow = A + (long long)i * K;
  for (int k = 0; k < K; ++k) {
    float av = arow[k];
    a0 += av * W[2 * k];
    a1 += av * W[2 * k + 1];
  }
  if (accum) { a0 += C[2 * i]; a1 += C[2 * i + 1]; }
  C[2 * i] = a0; C[2 * i + 1] = a1;
}

// ---------------- edge_index passthrough (as floats) ----------------
__global__ void k_ei(const long long* __restrict__ ei, float* __restrict__ dst, long long n) {
  long long i = (long long)blockIdx.x * blockDim.x + threadIdx.x;
  long long s = (long long)gridDim.x * blockDim.x;
  for (; i < n; i += s) dst[i] = (float)ei[i];
}

static inline unsigned gsz(long long n, int b) { return (unsigned)((n + b - 1) / b); }

extern "C" void kernel_launch(void* const* d_in, const int* in_sizes, int n_in,
                              void* d_out, int out_size, void* d_ws, size_t ws_size,
                              hipStream_t stream) {
  const float*     x  = (const float*)d_in[0];
  const long long* ei = (const long long*)d_in[1];
  const float* W1 = (const float*)d_in[2];
  const float* b1 = (const float*)d_in[3];
  const float* W2 = (const float*)d_in[4];
  const float* b2 = (const float*)d_in[5];
  const float* W3 = (const float*)d_in[6];
  const float* b3 = (const float*)d_in[7];
  float* out = (float*)d_out;

  const int N = in_sizes[0] / F_INC;
  const int E = in_sizes[1] / 2;
  const long long* row = ei;
  const long long* col = ei + E;

  // workspace layout (floats)
  float* p = (float*)d_ws;
  float* deg  = p; p += N;
  float* dinv = p; p += N;
  float* nrm  = p; p += E;
  float* Xp   = p; p += (long long)N * F_PADC;   // padded x  (layer-1 T0)
  float* Ta   = p; p += (long long)N * F_PADC;   // T1 buffer (reused each layer)
  float* Tb   = p; p += (long long)N * F_PADC;   // T2 buffer (reused each layer)
  float* Wp   = p; p += 3LL * F_PADC * HIDC;     // padded W1
  float* h1   = p; p += (long long)N * HIDC;
  float* h2   = p; p += (long long)N * HIDC;

  const long long nh  = (long long)N * HIDC;
  const long long npd = (long long)N * F_PADC;
  const int PROP_GRID = (E + 7) / 8;
  dim3 gW((N + 15) / 16, HIDC / 16);

  // ---- graph normalization ----
  k_zero<<<gsz(N, 256), 256, 0, stream>>>(deg, N);
  k_deg <<<gsz(E, 256), 256, 0, stream>>>(row, deg, E);
  k_dinv<<<gsz(N, 256), 256, 0, stream>>>(deg, dinv, N);
  k_norm<<<gsz(E, 256), 256, 0, stream>>>(row, col, dinv, nrm, E);

  // ---- padded operands for layer 1 ----
  k_pad_x<<<gsz(npd, 256), 256, 0, stream>>>(x, Xp, N);
  k_pad_w<<<gsz(3LL * F_PADC * HIDC, 256), 256, 0, stream>>>(W1, Wp);

  // ---- layer 1: h1 = relu6(T0@W0 + T1@W1 + T2@W2 + b1), T0 = x ----
  k_zero<<<gsz(nh, 256), 256, 0, stream>>>(h1, nh);
  k_wmma_gemm<<<gW, 32, 0, stream>>>(h1, Xp, Wp, N, F_PADC);
  k_zero<<<gsz(npd, 256), 256, 0, stream>>>(Ta, npd);
  k_prop<<<PROP_GRID, 128, 0, stream>>>(Ta, Xp, row, col, nrm, E, F_INC, F_PADC, 1.0f);
  k_wmma_gemm<<<gW, 32, 0, stream>>>(h1, Ta, Wp + (long long)F_PADC * HIDC, N, F_PADC);
  k_negcopy<<<gsz(npd, 256), 256, 0, stream>>>(Tb, Xp, npd);              // T2 = -T0 ...
  k_prop<<<PROP_GRID, 128, 0, stream>>>(Tb, Ta, row, col, nrm, E, F_INC, F_PADC, 2.0f); // ... + 2*prop(T1)
  k_wmma_gemm<<<gW, 32, 0, stream>>>(h1, Tb, Wp + 2LL * F_PADC * HIDC, N, F_PADC);
  k_bias_act<<<gsz(nh, 256), 256, 0, stream>>>(h1, b1, nh, HIDC, 1);

  // ---- layer 2 (stride = 128, reuse Ta/Tb) ----
  k_zero<<<gsz(nh, 256), 256, 0, stream>>>(h2, nh);
  k_wmma_gemm<<<gW, 32, 0, stream>>>(h2, h1, W2, N, HIDC);
  k_zero<<<gsz(nh, 256), 256, 0, stream>>>(Ta, nh);
  k_prop<<<PROP_GRID, 128, 0, stream>>>(Ta, h1, row, col, nrm, E, HIDC, HIDC, 1.0f);
  k_wmma_gemm<<<gW, 32, 0, stream>>>(h2, Ta, W2 + (long long)HIDC * HIDC, N, HIDC);
  k_negcopy<<<gsz(nh, 256), 256, 0, stream>>>(Tb, h1, nh);
  k_prop<<<PROP_GRID, 128, 0, stream>>>(Tb, Ta, row, col, nrm, E, HIDC, HIDC, 2.0f);
  k_wmma_gemm<<<gW, 32, 0, stream>>>(h2, Tb, W2 + 2LL * HIDC * HIDC, N, HIDC);
  k_bias_act<<<gsz(nh, 256), 256, 0, stream>>>(h2, b2, nh, HIDC, 1);

  // ---- layer 3 (out-dim 2: scalar GEMM) ----
  k_gemm2<<<gsz(N, 256), 256, 0, stream>>>(out, h2, W3, N, HIDC, 0);
  k_zero<<<gsz(nh, 256), 256, 0, stream>>>(Ta, nh);
  k_prop<<<PROP_GRID, 128, 0, stream>>>(Ta, h2, row, col, nrm, E, HIDC, HIDC, 1.0f);
  k_gemm2<<<gsz(N, 256), 256, 0, stream>>>(out, Ta, W3 + (long long)HIDC * NCLS, N, HIDC, 1);
  k_negcopy<<<gsz(nh, 256), 256, 0, stream>>>(Tb, h2, nh);
  k_prop<<<PROP_GRID, 128, 0, stream>>>(Tb, Ta, row, col, nrm, E, HIDC, HIDC, 2.0f);
  k_gemm2<<<gsz(N, 256), 256, 0, stream>>>(out, Tb, W3 + 2LL * HIDC * NCLS, N, HIDC, 1);
  k_bias_act<<<gsz((long long)N * NCLS, 256), 256, 0, stream>>>(out, b3, (long long)N * NCLS, NCLS, 0);

  // ---- second tuple element: edge_index copied (converted) into output tail ----
  long long tail = (long long)out_size - (long long)N * NCLS;
  if (tail > 0) {
    long long cnt = tail < (long long)2 * E ? tail : (long long)2 * E;
    k_ei<<<gsz(cnt, 256), 256, 0, stream>>>(ei, out + (long long)N * NCLS, cnt);
  }
}